// Attention_14663018349107
// MI455X (gfx1250) — compile-verified
//
#include <hip/hip_runtime.h>

// ---------------------------------------------------------------------------
// Cross-attention (B=1, T=4096, D=512, H=8, hd=64), fp32 end-to-end.
// All GEMMs use V_WMMA_F32_16X16X4_F32 (full-precision CDNA5 matrix path).
// fp32 WMMA operand layouts (ISA 7.12.2):
//   A (16x4): lane L holds row m=L&15; VGPR0/1 = K {0,1} (L<16) or {2,3} (L>=16)
//   B (4x16): lane L holds col n=L&15; VGPR0/1 = K {0,1} (L<16) or {2,3} (L>=16)
//   C/D (16x16): VGPR r, lane L -> M = r + 8*(L>=16), N = L&15
// ---------------------------------------------------------------------------

typedef __attribute__((ext_vector_type(2))) float v2f;
typedef __attribute__((ext_vector_type(8))) float v8f;

#define DIM    512
#define SEQ    4096
#define CTXIN  384
#define K2     768     // 2*CTXIN : conv(k=2,stride=2) as GEMM reduction dim
#define OCH    1024    // 2*DIM conv output channels
#define HEADS  8
#define HD     64

static __device__ __forceinline__ v8f wmma_f32(v2f a, v2f b, v8f c) {
  return __builtin_amdgcn_wmma_f32_16x16x4_f32(false, a, false, b, (short)0, c,
                                               false, false);
}

// ---------------------------------------------------------------------------
// C[M x N] = A[M x K] @ W[N x K]^T + bias.  One wave per 16x16 tile.
// Grid must supply exactly (M/16)*(N/16) waves (8 waves / 256-thread block).
// ---------------------------------------------------------------------------
__global__ __launch_bounds__(256) void gemm_awt_bias(
    const float* __restrict__ A, const float* __restrict__ W,
    const float* __restrict__ bias, float* __restrict__ C,
    int N, int K)
{
  const int lane  = threadIdx.x & 31;
  const int wave  = (blockIdx.x * blockDim.x + threadIdx.x) >> 5;
  const int tilesN = N >> 4;
  const int m0 = (wave / tilesN) << 4;
  const int n0 = (wave % tilesN) << 4;
  const int mrow = lane & 15;           // A row / B col / C col for this lane
  const int kb   = (lane >> 4) << 1;    // K sub-offset {0,2}

  const float* a = A + (size_t)(m0 + mrow) * K + kb;
  const float* w = W + (size_t)(n0 + mrow) * K + kb;

  v8f acc = {0.f, 0.f, 0.f, 0.f, 0.f, 0.f, 0.f, 0.f};
  for (int k = 0; k < K; k += 4) {
    v2f av; av.x = a[k]; av.y = a[k + 1];
    v2f bv; bv.x = w[k]; bv.y = w[k + 1];
    acc = wmma_f32(av, bv, acc);
  }

  const int   mo = (lane >> 4) << 3;    // C row base {0,8}
  const float bn = bias[n0 + mrow];
#pragma unroll
  for (int r = 0; r < 8; ++r)
    C[(size_t)(m0 + mo + r) * N + n0 + mrow] = acc[r] + bn;
}

// ---------------------------------------------------------------------------
// Conv1d(in=384,out=1024,k=2,stride=2) == ctx.view(4096,768) @ W2^T,
//   W2[c][j] = kv_w[c, j, 0]        (j < 384)
//            = kv_w[c, j-384, 1]    (j >= 384)          (kv_w is (1024,384,2))
// Epilogue replays torch's raw view (B,2D,T)->(B,T,2,D): flat i = c*4096+t,
//   q=i>>10, s=(i>>9)&1, d=i&511 ; adds kv_b + pos ; scatters to K / V.
// Grid: (4096/16)*(1024/16) = 16384 waves.
// ---------------------------------------------------------------------------
__global__ __launch_bounds__(256) void conv_kv(
    const float* __restrict__ ctx,   // (2*SEQ,384) == (SEQ,768) row-major
    const float* __restrict__ kvw,   // (1024,384,2)
    const float* __restrict__ kvb,   // (1024)
    const float* __restrict__ pos,   // (SEQ,DIM)
    float* __restrict__ Kb, float* __restrict__ Vb)   // (SEQ,DIM) each
{
  const int lane  = threadIdx.x & 31;
  const int wave  = (blockIdx.x * blockDim.x + threadIdx.x) >> 5;
  const int m0 = (wave >> 6) << 4;          // time tile   (tilesN = 1024/16 = 64)
  const int n0 = (wave & 63) << 4;          // channel tile
  const int mrow = lane & 15;
  const int kb   = (lane >> 4) << 1;

  const float* a    = ctx + (size_t)(m0 + mrow) * K2 + kb;
  const float* wrow = kvw + (size_t)(n0 + mrow) * K2;   // 768 floats: (384,2)

  v8f acc = {0.f, 0.f, 0.f, 0.f, 0.f, 0.f, 0.f, 0.f};
  for (int k = 0; k < K2; k += 4) {
    const int j0 = k + kb, j1 = j0 + 1;     // both on same side of 384 (k%4==0)
    v2f av; av.x = a[k]; av.y = a[k + 1];
    v2f bv;
    bv.x = (j0 < CTXIN) ? wrow[2 * j0] : wrow[2 * (j0 - CTXIN) + 1];
    bv.y = (j1 < CTXIN) ? wrow[2 * j1] : wrow[2 * (j1 - CTXIN) + 1];
    acc = wmma_f32(av, bv, acc);
  }

  const int   mo = (lane >> 4) << 3;
  const int   c  = n0 + mrow;
  const float bn = kvb[c];
#pragma unroll
  for (int r = 0; r < 8; ++r) {
    const int t = m0 + mo + r;
    const unsigned i = (unsigned)c * SEQ + (unsigned)t;
    const int q = i >> 10, sb = (i >> 9) & 1, d = i & 511;
    const float val = acc[r] + bn + pos[q * DIM + d];
    (sb ? Vb : Kb)[q * DIM + d] = val;
  }
}

// ---------------------------------------------------------------------------
// Flash attention, one wave per (head, 16-row Q tile): 8*256 = 2048 waves.
// S = Q Kt / 8 via WMMA; online softmax (row stats live per lane for rows
// r + 8*(lane>=16), reduced across the 16-lane half with shfl_xor 1/2/4/8);
// P converted D-layout -> A-layout through a private LDS tile; O += P V.
// No early exits -> EXEC all ones for every WMMA; all waves hit barriers.
// ---------------------------------------------------------------------------
__global__ __launch_bounds__(256) void flash_attn(
    const float* __restrict__ Qb, const float* __restrict__ Kb,
    const float* __restrict__ Vb, float* __restrict__ Ob)
{
  __shared__ float plds[8][16][17];         // per-wave P tile (+1 pad)

  const int lane = threadIdx.x & 31;
  const int wv   = threadIdx.x >> 5;
  const int gw   = blockIdx.x * 8 + wv;
  const int head = gw >> 8;
  const int q0   = (gw & 255) << 4;
  const int hoff = head << 6;               // head*64
  const int mrow = lane & 15;
  const int kb   = (lane >> 4) << 1;
  const int mo   = (lane >> 4) << 3;

  // Q tile as 16 pre-loaded A-fragments (hd = 64 = 16 k-steps of 4)
  v2f qf[16];
#pragma unroll
  for (int s = 0; s < 16; ++s) {
    const float* qp = Qb + (size_t)(q0 + mrow) * DIM + hoff + s * 4 + kb;
    qf[s].x = qp[0]; qf[s].y = qp[1];
  }

  v8f o[4];
#pragma unroll
  for (int nb = 0; nb < 4; ++nb) o[nb] = (v8f){0.f,0.f,0.f,0.f,0.f,0.f,0.f,0.f};
  float mrun[8], lrun[8];
#pragma unroll
  for (int r = 0; r < 8; ++r) { mrun[r] = -3.0e38f; lrun[r] = 0.f; }

  for (int kc0 = 0; kc0 < SEQ; kc0 += 16) {
    // ---- S tile = Q (16x64) @ K_chunk^T (64x16) ----
    v8f s = (v8f){0.f,0.f,0.f,0.f,0.f,0.f,0.f,0.f};
#pragma unroll
    for (int st = 0; st < 16; ++st) {
      const float* kp = Kb + (size_t)(kc0 + mrow) * DIM + hoff + st * 4 + kb;
      v2f bv; bv.x = kp[0]; bv.y = kp[1];
      s = wmma_f32(qf[st], bv, s);
    }

    // ---- online softmax over this 16-wide chunk ----
#pragma unroll
    for (int r = 0; r < 8; ++r) {
      float v = s[r] * 0.125f;              // 1/sqrt(64)
      float rmax = v;
      rmax = fmaxf(rmax, __shfl_xor(rmax, 1));
      rmax = fmaxf(rmax, __shfl_xor(rmax, 2));
      rmax = fmaxf(rmax, __shfl_xor(rmax, 4));
      rmax = fmaxf(rmax, __shfl_xor(rmax, 8));
      const float mn = fmaxf(mrun[r], rmax);
      const float p  = __expf(v - mn);
      float rsum = p;
      rsum += __shfl_xor(rsum, 1);
      rsum += __shfl_xor(rsum, 2);
      rsum += __shfl_xor(rsum, 4);
      rsum += __shfl_xor(rsum, 8);
      const float corr = __expf(mrun[r] - mn);
      lrun[r] = lrun[r] * corr + rsum;
      mrun[r] = mn;
#pragma unroll
      for (int nb = 0; nb < 4; ++nb) o[nb][r] *= corr;
      plds[wv][r + mo][mrow] = p;           // C-layout -> LDS
    }
    __syncthreads();

    // ---- O += P (16x16) @ V_chunk (16x64) ----
#pragma unroll
    for (int kc = 0; kc < 4; ++kc) {
      v2f pa;                               // A-frag of P from LDS
      pa.x = plds[wv][mrow][kc * 4 + kb];
      pa.y = plds[wv][mrow][kc * 4 + kb + 1];
#pragma unroll
      for (int nb = 0; nb < 4; ++nb) {
        const float* vp =
            Vb + (size_t)(kc0 + kc * 4 + kb) * DIM + hoff + nb * 16 + mrow;
        v2f bv; bv.x = vp[0]; bv.y = vp[DIM];
        o[nb] = wmma_f32(pa, bv, o[nb]);
      }
    }
    __syncthreads();
  }

  // ---- finalize: divide by row sums, write O ----
#pragma unroll
  for (int r = 0; r < 8; ++r) {
    const float inv = 1.0f / lrun[r];
    const int row = q0 + r + mo;
#pragma unroll
    for (int nb = 0; nb < 4; ++nb)
      Ob[(size_t)row * DIM + hoff + nb * 16 + mrow] = o[nb][r] * inv;
  }
}

// ---------------------------------------------------------------------------
extern "C" void kernel_launch(void* const* d_in, const int* in_sizes, int n_in,
                              void* d_out, int out_size, void* d_ws, size_t ws_size,
                              hipStream_t stream)
{
  const float* query   = (const float*)d_in[0];
  const float* context = (const float*)d_in[1];
  const float* pos     = (const float*)d_in[2];
  const float* q_w     = (const float*)d_in[3];
  const float* q_b     = (const float*)d_in[4];
  const float* kv_w    = (const float*)d_in[5];
  const float* kv_b    = (const float*)d_in[6];
  const float* out_w   = (const float*)d_in[7];
  const float* out_b   = (const float*)d_in[8];
  float* out = (float*)d_out;

  // Workspace: Q, K, V, O buffers, 8 MB each (32 MB total).
  float* Qb = (float*)d_ws;
  float* Kb = Qb + (size_t)SEQ * DIM;
  float* Vb = Kb + (size_t)SEQ * DIM;
  float* Ob = Vb + (size_t)SEQ * DIM;

  // 1) Q projection: (4096x512) @ (512x512)^T  -> 8192 waves
  gemm_awt_bias<<<1024, 256, 0, stream>>>(query, q_w, q_b, Qb, DIM, DIM);
  // 2) KV conv-as-GEMM + scrambled-view scatter + pos add -> 16384 waves
  conv_kv<<<2048, 256, 0, stream>>>(context, kv_w, kv_b, pos, Kb, Vb);
  // 3) Flash attention: 8 heads x 256 q-tiles -> 2048 waves
  flash_attn<<<256, 256, 0, stream>>>(Qb, Kb, Vb, Ob);
  // 4) Output projection -> d_out
  gemm_awt_bias<<<1024, 256, 0, stream>>>(Ob, out_w, out_b, out, DIM, DIM);
}